// ReadOut_6528350290208
// MI455X (gfx1250) — compile-verified
//
#include <hip/hip_runtime.h>

#define G_SEGS 4096
#define D_FEAT 128
#define ROWS_PER_WAVE 128
#define WAVES_PER_BLOCK 4

typedef __attribute__((ext_vector_type(2))) float v2f;
typedef __attribute__((ext_vector_type(8))) float v8f;

// starts[g] = lower_bound(batch_index, g); starts[G_SEGS] = n
__global__ void seg_bounds_kernel(const int* __restrict__ bidx, int n,
                                  int* __restrict__ starts) {
    int g = blockIdx.x * blockDim.x + threadIdx.x;
    if (g > G_SEGS) return;
    int lo = 0, hi = n;
    while (lo < hi) {
        int mid = (lo + hi) >> 1;
        if (bidx[mid] < g) lo = mid + 1; else hi = mid;
    }
    starts[g] = lo;
}

// Segment-sum via V_WMMA_F32_16X16X4_F32:
//   D[16 segs x 16 feats] += A[16 segs x 4 rows](one-hot) * B[4 rows x 16 feats]
__global__ __launch_bounds__(WAVES_PER_BLOCK * 32)
void segsum_wmma_kernel(const float* __restrict__ x,
                        const int* __restrict__ bidx,
                        float* __restrict__ sums, int n) {
    const int wave  = threadIdx.x >> 5;
    const int lane  = threadIdx.x & 31;
    const int chunk = blockIdx.x * WAVES_PER_BLOCK + wave;
    const int r0    = chunk * ROWS_PER_WAVE;
    if (r0 >= n) return;                        // wave-uniform exit
    const int rEnd  = min(r0 + ROWS_PER_WAVE, n);
    const int rMain = r0 + ((rEnd - r0) & ~3);  // full 4-row tiles

    const int m = lane & 15;   // A: segment row / B,D: feature column
    const int h = lane >> 4;   // K-half selector (k = 2h, 2h+1)

    const int gFirst = bidx[r0];
    const int gLast  = bidx[rEnd - 1];

    // windows of 16 consecutive graph ids; sorted ids => usually one pass
    for (int gBase = gFirst; gBase <= gLast; gBase += 16) {
        v8f acc[8] = {};   // 8 feature blocks x (16 segs x 16 feats)

        const int*   ip = bidx + r0 + 2 * h;
        const float* p0 = x + (size_t)(r0 + 2 * h) * D_FEAT + m;

        // ---- main body: no clamping, no divergence ----
        for (int r = r0; r < rMain; r += 4, ip += 4, p0 += 4 * D_FEAT) {
            const int i0 = ip[0];              // row r+2h
            const int i1 = ip[1];              // row r+2h+1
            v2f a;
            a.x = (i0 - gBase == m) ? 1.0f : 0.0f;
            a.y = (i1 - gBase == m) ? 1.0f : 0.0f;

            __builtin_prefetch(p0 + 16 * D_FEAT, 0, 1);  // speculative

            v2f b[8];                          // batch all 16 loads first
            #pragma unroll
            for (int cb = 0; cb < 8; ++cb) {
                b[cb].x = p0[cb * 16];
                b[cb].y = p0[D_FEAT + cb * 16];
            }
            #pragma unroll
            for (int cb = 0; cb < 8; ++cb)
                acc[cb] = __builtin_amdgcn_wmma_f32_16x16x4_f32(
                    false, a, false, b[cb], (short)0, acc[cb], false, false);
        }

        // ---- tail: 0-3 rows, masked through one-hot A ----
        if (rMain < rEnd) {
            const int r0h = rMain + 2 * h;
            const int r1h = r0h + 1;
            const int r0c = min(r0h, rEnd - 1);
            const int r1c = min(r1h, rEnd - 1);
            const int i0  = bidx[r0c];
            const int i1  = bidx[r1c];
            v2f a;
            a.x = (r0h < rEnd && i0 - gBase == m) ? 1.0f : 0.0f;
            a.y = (r1h < rEnd && i1 - gBase == m) ? 1.0f : 0.0f;

            const float* q0 = x + (size_t)r0c * D_FEAT + m;
            const float* q1 = x + (size_t)r1c * D_FEAT + m;
            v2f b[8];
            #pragma unroll
            for (int cb = 0; cb < 8; ++cb) {
                b[cb].x = q0[cb * 16];
                b[cb].y = q1[cb * 16];
            }
            #pragma unroll
            for (int cb = 0; cb < 8; ++cb)
                acc[cb] = __builtin_amdgcn_wmma_f32_16x16x4_f32(
                    false, a, false, b[cb], (short)0, acc[cb], false, false);
        }

        // ---- flush: D layout n = lane&15, M = vgpr + 8*(lane>=16) ----
        #pragma unroll
        for (int v = 0; v < 8; ++v) {
            const int g = gBase + v + 8 * h;
            if (g <= gLast) {
                float* dst = sums + (size_t)g * D_FEAT + m;
                #pragma unroll
                for (int cb = 0; cb < 8; ++cb) {
                    unsafeAtomicAdd(dst + cb * 16, acc[cb][v]);
                }
            }
        }
    }
}

__global__ void finalize_kernel(float* __restrict__ out,
                                const int* __restrict__ starts) {
    int idx = blockIdx.x * blockDim.x + threadIdx.x;
    if (idx >= G_SEGS * D_FEAT) return;
    int g = idx >> 7;  // / D_FEAT
    float cnt = (float)(starts[g + 1] - starts[g]);
    out[idx] = out[idx] / cnt;
}

extern "C" void kernel_launch(void* const* d_in, const int* in_sizes, int n_in,
                              void* d_out, int out_size, void* d_ws, size_t ws_size,
                              hipStream_t stream) {
    const float* x    = (const float*)d_in[0];
    const int*   bidx = (const int*)d_in[1];
    float*       out  = (float*)d_out;
    const int    n    = in_sizes[1];

    int* starts = (int*)d_ws;  // 4097 ints

    hipMemsetAsync(d_out, 0, (size_t)G_SEGS * D_FEAT * sizeof(float), stream);

    seg_bounds_kernel<<<(G_SEGS + 256) / 256, 256, 0, stream>>>(bidx, n, starts);

    const int chunks = (n + ROWS_PER_WAVE - 1) / ROWS_PER_WAVE;
    const int blocks = (chunks + WAVES_PER_BLOCK - 1) / WAVES_PER_BLOCK;
    segsum_wmma_kernel<<<blocks, WAVES_PER_BLOCK * 32, 0, stream>>>(x, bidx, out, n);

    finalize_kernel<<<(G_SEGS * D_FEAT + 255) / 256, 256, 0, stream>>>(out, starts);
}